// VectorQuantizer_50895362457926
// MI455X (gfx1250) — compile-verified
//
#include <hip/hip_runtime.h>
#include <hip/hip_bf16.h>

typedef float v2f __attribute__((ext_vector_type(2)));
typedef float v8f __attribute__((ext_vector_type(8)));

#define VQ_D 256
#define VQ_NE 1024
#define KSTEPS 64          // 256 / 4
#define NTILES 64          // 1024 / 16
#define WAVES_PER_WG 8
#define ROWS_PER_WG 128    // 8 waves * 16 rows
#define LDS_STRIDE 260     // 256 + 4 pad: 16B-aligned rows, conflict-free b64 reads
#define BETA 0.25f

// ---- kernel 1: wnorm[c] = sum_d W[c][d]^2  (1024 codes) ----
__global__ __launch_bounds__(256) void vq_wnorm_kernel(const float* __restrict__ W,
                                                       float* __restrict__ wnorm) {
    const int c = blockIdx.x * 256 + threadIdx.x;
    const float4* wr = (const float4*)(W + c * VQ_D);
    float s = 0.f;
#pragma unroll
    for (int i = 0; i < VQ_D / 4; ++i) {
        float4 v = wr[i];
        s += v.x * v.x + v.y * v.y + v.z * v.z + v.w * v.w;
    }
    wnorm[c] = s;
}

// ---- kernel 2: main VQ (distances via V_WMMA_F32_16X16X4_F32, argmin, gather, loss) ----
__global__ __launch_bounds__(256) void vq_main_kernel(const float* __restrict__ z,
                                                      const float* __restrict__ W,
                                                      const float* __restrict__ wnorm,
                                                      float* __restrict__ zq_out,
                                                      float* __restrict__ loss_out,
                                                      float* __restrict__ idx_out) {
    __shared__ float wtile[16 * LDS_STRIDE];

    const int tid  = threadIdx.x;
    const int lane = tid & 31;
    const int wave = tid >> 5;
    const int rowBase = (blockIdx.x * WAVES_PER_WG + wave) * 16;

    const int m  = lane & 15;          // A row / B col / C col within tile
    const int kb = (lane >> 4) * 2;    // lane-half k offset within a 4-wide k slice

    // ---- load A tile (16 rows x 256) into 128 VGPRs; accumulate ||z||^2 partials ----
    v2f a[KSTEPS];
    float zn = 0.f;
    {
        const float* zrow = z + (size_t)(rowBase + m) * VQ_D + kb;
#pragma unroll
        for (int s = 0; s < KSTEPS; ++s) {
            v2f v = *(const v2f*)(zrow + 4 * s);
            a[s] = v;
            zn += v.x * v.x + v.y * v.y;
        }
    }
    zn += __shfl_xor(zn, 16, 32);      // lane l now holds ||z_row(l&15)||^2

    float bestS[8];
    int   bestI[8];
#pragma unroll
    for (int r = 0; r < 8; ++r) { bestS[r] = 3.4e38f; bestI[r] = 0; }

    // ---- loop over 64 code tiles of 16 codes each ----
    for (int t = 0; t < NTILES; ++t) {
        const int codeBase = t * 16;

        __syncthreads();               // protect wtile from previous iteration's readers
        {   // cooperative stage of 16x256 W tile into padded LDS
            const int r  = tid >> 4;           // 0..15 code row
            const int dd = (tid & 15) * 16;    // 0..240 dim offset
            const float* src = W + (size_t)(codeBase + r) * VQ_D + dd;
            float* dst = wtile + r * LDS_STRIDE + dd;
#pragma unroll
            for (int q = 0; q < 4; ++q)
                *(float4*)(dst + 4 * q) = *(const float4*)(src + 4 * q);
        }
        __syncthreads();

        v8f c = {0.f, 0.f, 0.f, 0.f, 0.f, 0.f, 0.f, 0.f};
        const float* bptr = wtile + m * LDS_STRIDE + kb;
#pragma unroll
        for (int s = 0; s < KSTEPS; ++s) {
            v2f b = *(const v2f*)(bptr + 4 * s);
            c = __builtin_amdgcn_wmma_f32_16x16x4_f32(
                    /*neg_a=*/false, a[s], /*neg_b=*/false, b,
                    /*c_mod=*/(short)0, c, /*reuse_a=*/false, /*reuse_b=*/false);
        }

        // score = ||w||^2 - 2 z.w ; per-lane running argmin (C layout: M = r + 8*(lane>=16))
        const float wn = wnorm[codeBase + m];
        const int   ci = codeBase + m;
#pragma unroll
        for (int r = 0; r < 8; ++r) {
            float sc = __builtin_fmaf(-2.f, c[r], wn);
            if (sc < bestS[r]) { bestS[r] = sc; bestI[r] = ci; }
        }
    }

    // ---- 16-lane butterfly argmin (keep halves separate: they hold different rows) ----
#pragma unroll
    for (int r = 0; r < 8; ++r) {
#pragma unroll
        for (int off = 8; off >= 1; off >>= 1) {
            float os = __shfl_xor(bestS[r], off, 16);
            int   oi = __shfl_xor(bestI[r], off, 16);
            if (os < bestS[r] || (os == bestS[r] && oi < bestI[r])) {
                bestS[r] = os; bestI[r] = oi;
            }
        }
    }

    // ---- emit: z_q (codebook row copy), loss, idx for the wave's 16 rows ----
#pragma unroll
    for (int t2 = 0; t2 < 16; ++t2) {
        const int r = t2 & 7, h = t2 >> 3;          // row = rowBase + r + 8h = rowBase + t2
        const float bs  = __shfl(bestS[r], h * 16, 32);
        const int   bi  = __shfl(bestI[r], h * 16, 32);
        const float znr = __shfl(zn, t2, 32);
        const int   row = rowBase + t2;

        if (lane == 0) {
            loss_out[row] = (1.0f + BETA) * (znr + bs) * (1.0f / (float)VQ_D);
            idx_out[row]  = (float)bi;
        }
        // all 32 lanes copy W[bi][0..255] -> zq_out[row][0..255] (64 x float4)
        const float4* src = (const float4*)(W + (size_t)bi * VQ_D) + lane;
        float4*       dst = (float4*)(zq_out + (size_t)row * VQ_D) + lane;
        dst[0]  = src[0];
        dst[32] = src[32];
    }
}

extern "C" void kernel_launch(void* const* d_in, const int* in_sizes, int n_in,
                              void* d_out, int out_size, void* d_ws, size_t ws_size,
                              hipStream_t stream) {
    const float* z = (const float*)d_in[0];   // [32,2048,256] f32
    const float* W = (const float*)d_in[1];   // [1024,256]    f32

    const int N = in_sizes[0] / VQ_D;         // 65536 rows

    float* zq   = (float*)d_out;              // [N,256]
    float* loss = zq + (size_t)N * VQ_D;      // [N]
    float* idxf = loss + N;                   // [N] (as float)

    float* wnorm = (float*)d_ws;              // 1024 floats of scratch

    vq_wnorm_kernel<<<VQ_NE / 256, 256, 0, stream>>>(W, wnorm);
    vq_main_kernel<<<N / ROWS_PER_WG, 256, 0, stream>>>(z, W, wnorm, zq, loss, idxf);
}